// HamiltonianClassifier_11355893531272
// MI455X (gfx1250) — compile-verified
//
#include <hip/hip_runtime.h>
#include <hip/hip_bf16.h>

// HamiltonianClassifier, MI455X (gfx1250, wave32).
//
// Math reduction: score[b] = Re(e^T H conj(e)) with H = X^T X / 128 (X real)
//               = (||X er||^2 + ||X ei||^2) / 128
// => never build the 768x768 Hamiltonian. Two passes over x (12.6 MB; the
// second pass hits the 192 MB L2), a tiny LDS state-vector circuit, and a
// per-batch skinny GEMM C(128x16) = X(128x768) @ [er|ei|0...] driven by
// V_WMMA_F32_16X16X4_F32 (full f32 precision).

#define BATCH 32
#define SENT  128
#define EMB   768
#define DIM   1024
#define NW    10

typedef __attribute__((ext_vector_type(2))) float v2f;
typedef __attribute__((ext_vector_type(8))) float v8f;

// ---------------------------------------------------------------------------
// Kernel 1: fused mean-over-sentence + 10-qubit circuit, one block per batch.
// State (1024 complex) lives in LDS; 512 threads = one amplitude pair each.
// Writes e (interleaved complex64) to d_out+32 and er/ei to workspace.
// ---------------------------------------------------------------------------
__global__ __launch_bounds__(512) void circuit_kernel(
    const float* __restrict__ x, const float* __restrict__ tx,
    const float* __restrict__ ty, float* __restrict__ out,
    float* __restrict__ wer, float* __restrict__ wei) {
  __shared__ float sre[DIM];
  __shared__ float sim[DIM];
  __shared__ float sg[4 * NW];  // cx[10], sx[10], cy[10], sy[10]

  const int b = blockIdx.x;
  const int t = threadIdx.x;

  if (t < NW) {
    float ax = tx[t], ay = ty[t];
    sg[t]          = __cosf(ax);
    sg[NW + t]     = __sinf(ax);
    sg[2 * NW + t] = __cosf(ay);
    sg[3 * NW + t] = __sinf(ay);
  }

  // Mean over sentence axis -> real initial state, zero-padded to 1024.
  for (int i = t; i < DIM; i += 512) {
    float acc = 0.0f;
    if (i < EMB) {
      const float* p = x + (size_t)b * SENT * EMB + i;
#pragma unroll 8
      for (int s = 0; s < SENT; ++s) acc += p[s * EMB];
      acc *= (1.0f / SENT);
    }
    sre[i] = acc;
    sim[i] = 0.0f;
  }
  __syncthreads();

  // 3 x (RX layer, RY layer, CZ-ring sign). Wire k acts on bit (9-k)
  // (wire 0 = MSB in the kron ordering of the reference).
  for (int rep = 0; rep < 3; ++rep) {
    // RX: gate [[c, -i s], [-i s, c]]
    for (int k = 0; k < NW; ++k) {
      const int bpos = (NW - 1) - k;
      const int m = 1 << bpos;
      const int lo = ((t >> bpos) << (bpos + 1)) | (t & (m - 1));
      const int hi = lo | m;
      const float c = sg[k], s = sg[NW + k];
      float ar = sre[lo], ai = sim[lo], br = sre[hi], bi = sim[hi];
      sre[lo] = c * ar + s * bi;
      sim[lo] = c * ai - s * br;
      sre[hi] = c * br + s * ai;
      sim[hi] = c * bi - s * ar;
      __syncthreads();
    }
    // RY: gate [[c, -s], [s, c]] (real)
    for (int k = 0; k < NW; ++k) {
      const int bpos = (NW - 1) - k;
      const int m = 1 << bpos;
      const int lo = ((t >> bpos) << (bpos + 1)) | (t & (m - 1));
      const int hi = lo | m;
      const float c = sg[2 * NW + k], s = sg[3 * NW + k];
      float ar = sre[lo], ai = sim[lo], br = sre[hi], bi = sim[hi];
      sre[lo] = c * ar - s * br;
      sim[lo] = c * ai - s * bi;
      sre[hi] = s * ar + c * br;
      sim[hi] = s * ai + c * bi;
      __syncthreads();
    }
    // CZ ring + CZ(0,9): diagonal sign = (-1)^(adjacent-bit-pair parity + wrap)
    for (int i = t; i < DIM; i += 512) {
      int par = __popc(i & (i >> 1)) + ((i >> 9) & i & 1);
      if (par & 1) {
        sre[i] = -sre[i];
        sim[i] = -sim[i];
      }
    }
    __syncthreads();
  }

  // Emit e (interleaved complex64) + er/ei scratch for the score GEMM.
  float* oute = out + BATCH;
  for (int i = t; i < DIM; i += 512) {
    float r = sre[i], m2 = sim[i];
    size_t o = ((size_t)b * DIM + i) * 2;
    oute[o]     = r;
    oute[o + 1] = m2;
    if (i < EMB) {
      wer[b * EMB + i] = r;
      wei[b * EMB + i] = m2;
    }
  }
}

// ---------------------------------------------------------------------------
// Kernel 2: per-batch skinny GEMM via V_WMMA_F32_16X16X4_F32.
// One block (8 wave32s) per batch; wave w owns C rows [16w, 16w+16).
// A fragment layout (ISA 16x4 f32 A): lane l (0..15) row M=l; lane half
// selects K pair {0,1} vs {2,3}.
//
// B fragments come from LDS laid out as [er(768) | ei(768) | zeros(768)]:
// lane 0 bases at er, lane 1 at ei, every other lane at the zero region.
// All lanes then share the SAME per-iteration offset (+k), so the k term
// folds into the ds_load_b64 immediate-offset field and the inner loop is
// exactly {global_load_b64 imm, ds_load_b64 imm, v_wmma}, pipelined one
// iteration ahead so loads fly under the matrix op.
//
// score[b] = sum over rows of (C[:,0]^2 + C[:,1]^2) / 128, clipped to [0,1].
// ---------------------------------------------------------------------------
__global__ __launch_bounds__(256) void score_kernel(
    const float* __restrict__ x, const float* __restrict__ wer,
    const float* __restrict__ wei, float* __restrict__ out) {
  __shared__ float sbuf[3 * EMB];  // [0,768): er, [768,1536): ei, [1536,2304): 0
  __shared__ float wpart[8];

  const int b = blockIdx.x;
  const int t = threadIdx.x;

  for (int i = t; i < EMB; i += 256) {
    sbuf[i]           = wer[b * EMB + i];
    sbuf[EMB + i]     = wei[b * EMB + i];
    sbuf[2 * EMB + i] = 0.0f;
  }
  __syncthreads();

  const int wave = t >> 5;
  const int lane = t & 31;
  const int half = lane >> 4;        // K sub-pair select
  const int l    = lane & 15;        // A: row-in-tile; B/C: column
  const int row  = (wave << 4) | l;  // sentence position 0..127

  const v2f* arow = (const v2f*)(x + ((size_t)b * SENT + row) * EMB);

  // Per-lane constant B base; uniform +k stride for every lane.
  const int bbase = (l == 0) ? 0 : (l == 1) ? EMB : 2 * EMB;
  const float* bp = sbuf + bbase + 2 * half;

  v8f acc = {};
  v2f a  = arow[half];            // K = {2h, 2h+1}
  v2f bb = *(const v2f*)bp;

#pragma unroll 4
  for (int k = 0; k < EMB - 4; k += 4) {
    v2f a_n  = arow[((k + 4) >> 1) + half];
    v2f bb_n = *(const v2f*)(bp + k + 4);
    acc = __builtin_amdgcn_wmma_f32_16x16x4_f32(
        /*neg_a=*/false, a, /*neg_b=*/false, bb,
        /*c_mod=*/(short)0, acc, /*reuse_a=*/false, /*reuse_b=*/false);
    a  = a_n;
    bb = bb_n;
  }
  acc = __builtin_amdgcn_wmma_f32_16x16x4_f32(
      false, a, false, bb, (short)0, acc, false, false);

  // C layout: lanes 0-15 hold (M=j, N=lane) in acc[j]; lanes 16-31 hold
  // (M=8+j, N=lane-16). Columns 0/1 carry X@er and X@ei -> sum of squares.
  float local = 0.0f;
  if (l < 2) {
#pragma unroll
    for (int j = 0; j < 8; ++j) local += acc[j] * acc[j];
  }
  for (int off = 16; off > 0; off >>= 1) local += __shfl_xor(local, off, 32);
  if (lane == 0) wpart[wave] = local;
  __syncthreads();

  if (t == 0) {
    float tot = 0.0f;
#pragma unroll
    for (int w = 0; w < 8; ++w) tot += wpart[w];
    tot *= (1.0f / SENT);
    out[b] = fminf(fmaxf(tot, 0.0f), 1.0f);
  }
}

extern "C" void kernel_launch(void* const* d_in, const int* in_sizes, int n_in,
                              void* d_out, int out_size, void* d_ws, size_t ws_size,
                              hipStream_t stream) {
  const float* x  = (const float*)d_in[0];
  const float* tx = (const float*)d_in[1];
  const float* ty = (const float*)d_in[2];
  float* out = (float*)d_out;          // [0..31] scores, then e interleaved
  float* wer = (float*)d_ws;           // 32*768 floats: Re(e)[:768]
  float* wei = wer + BATCH * EMB;      // 32*768 floats: Im(e)[:768]

  circuit_kernel<<<BATCH, 512, 0, stream>>>(x, tx, ty, out, wer, wei);
  score_kernel<<<BATCH, 256, 0, stream>>>(x, wer, wei, out);
}